// NCE_20040317403236
// MI455X (gfx1250) — compile-verified
//
#include <hip/hip_runtime.h>
#include <math.h>

// CDNA5 / gfx1250 NCE objective.
//
// Reformulation: with c = cos(a), s = sin(a) per node,
//   lp[k,smp] = 0.5*(c^T P_k c + s^T P_k s) + s^T Q_k c
// where P_k[i][j] = P_k[j][i] = theta0[k, z(i,j)]  (symmetric, zero diag)
//       Q_k[i][j] = theta1[k,z], Q_k[j][i] = -theta1[k,z] (antisymmetric).
// This is pure dense GEMM work for the WMMA pipe (f32 16x16x4), with the
// only transcendentals being sincos of the raw 8192x128 phases.
//
// Stage 0: build P/Q (1 MB scratch, L2-resident).
// Stage 1: per-wave 16-sample tile; for each k: 8 j-tiles x 32 K-steps of
//          3x V_WMMA_F32_16X16X4_F32 (A from LDS, B coalesced from L2),
//          quadratic-form dot in the tile epilogue, wave32 __shfl_xor
//          reduction, logsumexp over K in-lane.
// Stage 2: deterministic single-block tree reduction -> scalar objective.

typedef __attribute__((ext_vector_type(2))) float v2f;
typedef __attribute__((ext_vector_type(8))) float v8f;

#define NODES 128
#define KMIX  8
#define ZP    (NODES * (NODES - 1) / 2)   // 8128
#define MATSZ (NODES * NODES)             // 16384 floats per matrix

__device__ __forceinline__ int zstart_row(int i) {
    // first z index of row i in triu_indices(NODES, k=1) order
    return (255 * i - i * i) / 2;         // NODES==128
}

// ---------------------------------------------------------------------------
// Stage 0: expand theta (K,2,Z) into PQ = [P_0,Q_0, P_1,Q_1, ...], each
// 128x128, stored so the GEMM B-operand reads are row-contiguous:
//   P mem[j*128+i] = theta0 (symmetric)   Q mem[j*128+i] = Q[i][j]
// Diagonals are explicitly zeroed (d_ws is poisoned by the harness).
// ---------------------------------------------------------------------------
__global__ __launch_bounds__(256) void nce_build_pq(
    const float* __restrict__ theta, float* __restrict__ PQ)
{
    int idx = blockIdx.x * 256 + threadIdx.x;
    const int totalPairs = KMIX * ZP;
    if (idx < totalPairs) {
        int k = idx / ZP;
        int z = idx - k * ZP;
        // invert z -> (i,j), i<j
        int i = (int)((255.0 - sqrt(255.0 * 255.0 - 8.0 * (double)z)) * 0.5);
        if (i < 0) i = 0;
        if (i > 126) i = 126;
        while (i < 126 && zstart_row(i + 1) <= z) ++i;
        while (i > 0 && zstart_row(i) > z) --i;
        int j = i + 1 + (z - zstart_row(i));

        float t0 = theta[(size_t)k * 2 * ZP + z];        // theta0[k,z]
        float t1 = theta[(size_t)k * 2 * ZP + ZP + z];   // theta1[k,z]
        float* Pk = PQ + (size_t)(2 * k) * MATSZ;
        float* Qk = Pk + MATSZ;
        // P symmetric: mem[a*128+b] = P[b][a] = t0 for both orders
        Pk[i * NODES + j] = t0;
        Pk[j * NODES + i] = t0;
        // Q: mem[a*128+b] must equal Q[b][a]; Q[i][j]=+t1, Q[j][i]=-t1
        Qk[j * NODES + i] = t1;     // = Q[i][j]
        Qk[i * NODES + j] = -t1;    // = Q[j][i]
    } else {
        int d = idx - totalPairs;   // zero the diagonals
        if (d < KMIX * 2 * NODES) {
            int mat = d / NODES, ii = d - mat * NODES;
            PQ[(size_t)mat * MATSZ + ii * NODES + ii] = 0.0f;
        }
    }
}

// ---------------------------------------------------------------------------
// Stage 1: 64 threads = 2 waves per block; each wave owns 16 samples
// (the 16 M-rows of its WMMA tiles).
//
// WMMA f32 16x16x4 operand layouts (ISA 7.12.2):
//   A 16x4 : lane l -> row m = l&15; VGPR0/1 = K slots {0,1} (lanes 0-15)
//            or {2,3} (lanes 16-31)  => per-lane kb = 4*ks + 2*(l>>4)
//   B 4x16 : lane l -> col n = l&15, same K-slot split
//   D 16x16: VGPR r, lanes 0-15 -> (M=r, N=lane); lanes 16-31 -> (M=r+8).
// ---------------------------------------------------------------------------
__global__ __launch_bounds__(64) void nce_lp_kernel(
    const float* __restrict__ X,
    const float* __restrict__ noise,
    const float* __restrict__ PQ,      // [K][2][128][128] as built above
    const float* __restrict__ logc,    // (K,)
    float* __restrict__ lse_out,       // (2*nN,)
    int nN)
{
    // padded rows: stride 132 dwords == 4 banks mod 64 -> conflict-free
    __shared__ float s_cos[32][NODES + 4];
    __shared__ float s_sin[32][NODES + 4];
    __shared__ float s_lp[32][KMIX + 1];

    const int tid    = threadIdx.x;
    const int lane   = tid & 31;
    const int wave   = tid >> 5;
    const int totalS = 2 * nN;
    const int sBase  = blockIdx.x * 32;

    // --- cos/sin of raw phases: the only transcendentals in the pipeline ---
    for (int idx = tid; idx < 32 * NODES; idx += 64) {
        int sl   = idx >> 7;            // NODES == 128
        int node = idx & (NODES - 1);
        int g    = sBase + sl;
        int gc   = (g < totalS) ? g : (totalS - 1);   // clamp keeps EXEC full
        const float* src = (gc < nN)
            ? (X     + (size_t)gc        * NODES + node)
            : (noise + (size_t)(gc - nN) * NODES + node);
        float v = *src;
        float sv, cv;
        sincosf(v, &sv, &cv);
        s_cos[sl][node] = cv;
        s_sin[sl][node] = sv;
    }
    __syncthreads();

    const int half = lane >> 4;         // K-slot half (0 -> kk{0,1}, 1 -> kk{2,3})
    const int n    = lane & 15;         // A row (sample) / B col (node i)
    const int slA  = wave * 16 + n;     // this lane's A-operand sample
    const float* rowc = &s_cos[slA][0];
    const float* rows = &s_sin[slA][0];

    for (int k = 0; k < KMIX; ++k) {
        const float* Pk = PQ + (size_t)(2 * k) * MATSZ;
        const float* Qk = Pk + MATSZ;

        float part[8];
#pragma unroll
        for (int r = 0; r < 8; ++r) part[r] = 0.0f;

        for (int jt = 0; jt < 8; ++jt) {
            const int jbase = jt * 16;
            v8f D1 = {}, D2 = {}, D3 = {};
            // B pointers: mem[kk*128 + i]; lanes 0-15 read contiguous i
            const float* Pb = Pk + (2 * half) * NODES + jbase + n;
            const float* Qb = Qk + (2 * half) * NODES + jbase + n;

            for (int ks = 0; ks < 32; ++ks) {
                const int kb = (ks << 2) + (half << 1);
                v2f ac, as2, bp, bq;
                ac.x  = rowc[kb];     ac.y  = rowc[kb + 1];   // ds_load_b64
                as2.x = rows[kb];     as2.y = rows[kb + 1];   // ds_load_b64
                bp.x  = Pb[0];        bp.y  = Pb[NODES];      // 2x coalesced b32
                bq.x  = Qb[0];        bq.y  = Qb[NODES];
                Pb += 4 * NODES;
                Qb += 4 * NODES;
                D1 = __builtin_amdgcn_wmma_f32_16x16x4_f32(
                        false, ac,  false, bp, (short)0, D1, false, false);
                D2 = __builtin_amdgcn_wmma_f32_16x16x4_f32(
                        false, as2, false, bp, (short)0, D2, false, false);
                D3 = __builtin_amdgcn_wmma_f32_16x16x4_f32(
                        false, ac,  false, bq, (short)0, D3, false, false);
            }
            // quadratic-form dot for this 16-column tile:
            // D1 = (P c)[m, j], D2 = (P s)[m, j], D3 = (Q c)[m, j]
#pragma unroll
            for (int r = 0; r < 8; ++r) {
                int sm = wave * 16 + r + 8 * half;   // D-row m = r + 8*half
                float cc = s_cos[sm][jbase + n];
                float sc = s_sin[sm][jbase + n];
                part[r] += 0.5f * (D1[r] * cc + D2[r] * sc) + D3[r] * sc;
            }
        }

        // reduce over the 16 lanes of each half (wave32 shuffles)
#pragma unroll
        for (int r = 0; r < 8; ++r) {
            float v = part[r];
            v += __shfl_xor(v, 1, 16);
            v += __shfl_xor(v, 2, 16);
            v += __shfl_xor(v, 4, 16);
            v += __shfl_xor(v, 8, 16);
            part[r] = v;
        }
        if ((lane & 15) == 0) {
#pragma unroll
            for (int r = 0; r < 8; ++r)
                s_lp[wave * 16 + r + 8 * half][k] = part[r];
        }
    }
    __syncthreads();

    // per-sample logsumexp over K (one thread per local sample)
    if (tid < 32) {
        float lp[KMIX];
#pragma unroll
        for (int k = 0; k < KMIX; ++k) lp[k] = s_lp[tid][k] + logc[k];
        float mx = lp[0];
#pragma unroll
        for (int k = 1; k < KMIX; ++k) mx = fmaxf(mx, lp[k]);
        float sum = 0.0f;
#pragma unroll
        for (int k = 0; k < KMIX; ++k) sum += expf(lp[k] - mx);
        float lse = mx + logf(sum);
        int g = sBase + tid;
        if (g < totalS) lse_out[g] = lse;
    }
}

// ---------------------------------------------------------------------------
// Stage 2: deterministic reduction (single block, fixed-order tree).
// J = mean(logN + lp_data - logaddexp(logN+lp_data, logM))
//   + mean(logN - logaddexp(logN+lp_noise, logM)),  logM == logN here.
// ---------------------------------------------------------------------------
__global__ __launch_bounds__(256) void nce_reduce_kernel(
    const float* __restrict__ lse, float* __restrict__ out, int nN)
{
    __shared__ float sA[256];
    __shared__ float sB[256];
    const int tid = threadIdx.x;
    const float logN = logf((float)nN);

    float accA = 0.0f, accB = 0.0f;
    for (int i = tid; i < nN; i += 256) {
        float a  = logN + lse[i];
        float m  = fmaxf(a, logN);
        float la = m + log1pf(expf(fminf(a, logN) - m));
        accA += a - la;
        float b  = logN + lse[nN + i];
        float m2 = fmaxf(b, logN);
        float lb = m2 + log1pf(expf(fminf(b, logN) - m2));
        accB += logN - lb;
    }
    sA[tid] = accA;
    sB[tid] = accB;
    __syncthreads();

    for (int s = 128; s > 0; s >>= 1) {
        if (tid < s) {
            sA[tid] += sA[tid + s];
            sB[tid] += sB[tid + s];
        }
        __syncthreads();
    }
    if (tid == 0)
        out[0] = sA[0] / (float)nN + sB[0] / (float)nN;
}

// ---------------------------------------------------------------------------
extern "C" void kernel_launch(void* const* d_in, const int* in_sizes, int n_in,
                              void* d_out, int out_size, void* d_ws, size_t ws_size,
                              hipStream_t stream)
{
    const float* X     = (const float*)d_in[0];
    const float* noise = (const float*)d_in[1];
    const float* theta = (const float*)d_in[2];
    const float* logc  = (const float*)d_in[3];

    const int nN = in_sizes[0] / NODES;              // 4096

    float* PQ  = (float*)d_ws;                       // K*2*16384 floats = 1 MB
    float* lse = PQ + (size_t)KMIX * 2 * MATSZ;      // 2*nN floats (32 KB)

    const int pqItems  = KMIX * ZP + KMIX * 2 * NODES;
    const int pqBlocks = (pqItems + 255) / 256;
    hipLaunchKernelGGL(nce_build_pq, dim3(pqBlocks), dim3(256), 0, stream,
                       theta, PQ);

    const int blocks = (2 * nN + 31) / 32;           // 32 samples / block
    hipLaunchKernelGGL(nce_lp_kernel, dim3(blocks), dim3(64), 0, stream,
                       X, noise, PQ, logc, lse, nN);

    hipLaunchKernelGGL(nce_reduce_kernel, dim3(1), dim3(256), 0, stream,
                       lse, (float*)d_out, nN);
}